// SpatialTransformer_69226282877327
// MI455X (gfx1250) — compile-verified
//
#include <hip/hip_runtime.h>

typedef float v4f __attribute__((ext_vector_type(4)));

// Problem constants (fixed by setup_inputs): src [2,2,128,160,128] f32, flow [2,3,128,160,128] f32
constexpr int Bn = 2, Cn = 2, Dn = 128, Hn = 160, Wn = 128;
constexpr int Sn  = Dn * Hn * Wn;  // 2,621,440 spatial points per batch
constexpr int HWn = Hn * Wn;       // 20,480

__device__ __forceinline__ float lerpf(float a, float b, float t) {
    return __builtin_fmaf(t, b - a, a);
}

// Trilinear sample (zeros padding) of both channels at one voxel coordinate.
// s0 = channel-0 base of this batch, s1 = s0 + Sn (channel 1).
__device__ __forceinline__ void sample_point(const float* __restrict__ s0,
                                             const float* __restrict__ s1,
                                             float iz, float iy, float ix,
                                             float& o0, float& o1) {
    float zf = floorf(iz), yf = floorf(iy), xf = floorf(ix);
    float tz = iz - zf, ty = iy - yf, tx = ix - xf;
    int z0 = (int)zf, y0 = (int)yf, x0 = (int)xf;

    // z0 in [0, Dn-2] etc. <=> all 8 corners in bounds
    bool interior = ((unsigned)z0 < (unsigned)(Dn - 1)) &
                    ((unsigned)y0 < (unsigned)(Hn - 1)) &
                    ((unsigned)x0 < (unsigned)(Wn - 1));
    if (interior) {
        int lin = (z0 * Hn + y0) * Wn + x0;
        const float* p = s0 + lin;   // single base per channel; corner offsets are
        const float* q = s1 + lin;   // compile-time immediates (max 82KB < 24-bit IOFFSET)
        // channel 0: lerp x, then y, then z (weights shared with channel 1)
        float c00 = lerpf(p[0],        p[1],            tx);
        float c01 = lerpf(p[Wn],       p[Wn + 1],       tx);
        float c10 = lerpf(p[HWn],      p[HWn + 1],      tx);
        float c11 = lerpf(p[HWn + Wn], p[HWn + Wn + 1], tx);
        o0 = lerpf(lerpf(c00, c01, ty), lerpf(c10, c11, ty), tz);
        // channel 1
        float d00 = lerpf(q[0],        q[1],            tx);
        float d01 = lerpf(q[Wn],       q[Wn + 1],       tx);
        float d10 = lerpf(q[HWn],      q[HWn + 1],      tx);
        float d11 = lerpf(q[HWn + Wn], q[HWn + Wn + 1], tx);
        o1 = lerpf(lerpf(d00, d01, ty), lerpf(d10, d11, ty), tz);
    } else {
        // Border: branchless masked 8-corner accumulation == zeros padding.
        // Clamp the index (always a legal address) and zero the weight when OOB.
        float a0 = 0.0f, a1 = 0.0f;
        float wz[2] = {1.0f - tz, tz};
        float wy[2] = {1.0f - ty, ty};
        float wx[2] = {1.0f - tx, tx};
#pragma unroll
        for (int dz = 0; dz < 2; ++dz)
#pragma unroll
            for (int dy = 0; dy < 2; ++dy)
#pragma unroll
                for (int dx = 0; dx < 2; ++dx) {
                    int zc = z0 + dz, yc = y0 + dy, xc = x0 + dx;
                    bool v = ((unsigned)zc < (unsigned)Dn) &
                             ((unsigned)yc < (unsigned)Hn) &
                             ((unsigned)xc < (unsigned)Wn);
                    int zq = min(max(zc, 0), Dn - 1);
                    int yq = min(max(yc, 0), Hn - 1);
                    int xq = min(max(xc, 0), Wn - 1);
                    int lin = (zq * Hn + yq) * Wn + xq;
                    float w = v ? (wz[dz] * wy[dy] * wx[dx]) : 0.0f;
                    a0 = __builtin_fmaf(w, s0[lin], a0);
                    a1 = __builtin_fmaf(w, s1[lin], a1);
                }
        o0 = a0;
        o1 = a1;
    }
}

// Grid: x = blockIdx.x -> octet of rows (y), y = blockIdx.y -> z slice, z = blockIdx.z -> batch.
// Block: 256 threads = 8 wave32; each wave covers one full row of 128 x (4 pts/thread).
__global__ __launch_bounds__(256)
void SpatialTransformer_warp3d_kernel(const float* __restrict__ src,
                                      const float* __restrict__ flow,
                                      float* __restrict__ out) {
    const int x = (threadIdx.x & 31) << 2;           // 0..124, step 4
    const int y = (blockIdx.x << 3) + (threadIdx.x >> 5);
    const int z = blockIdx.y;
    const int b = blockIdx.z;
    const int s = z * HWn + y * Wn + x;

    // flow[b, {z,y,x}, z, y, x]: three streaming (non-temporal) 16B loads.
    // NT keeps the 63MB flow stream from evicting the 42MB src set in L2.
    const float* fb = flow + (size_t)b * 3 * Sn + s;
    v4f fz = __builtin_nontemporal_load((const v4f*)fb);
    v4f fy = __builtin_nontemporal_load((const v4f*)(fb + Sn));
    v4f fx = __builtin_nontemporal_load((const v4f*)(fb + 2 * Sn));

    const float* s0 = src + (size_t)b * Cn * Sn;     // channel 0 base
    const float* s1 = s0 + Sn;                       // channel 1 base

    v4f o0, o1;
#pragma unroll
    for (int j = 0; j < 4; ++j) {
        float iz = (float)z + fz[j];
        float iy = (float)y + fy[j];
        float ix = (float)(x + j) + fx[j];
        float r0, r1;
        sample_point(s0, s1, iz, iy, ix, r0, r1);
        o0[j] = r0;
        o1[j] = r1;
    }

    // Streaming non-temporal 16B stores (never re-read).
    float* ob = out + (size_t)b * Cn * Sn + s;
    __builtin_nontemporal_store(o0, (v4f*)ob);
    __builtin_nontemporal_store(o1, (v4f*)(ob + Sn));
}

extern "C" void kernel_launch(void* const* d_in, const int* in_sizes, int n_in,
                              void* d_out, int out_size, void* d_ws, size_t ws_size,
                              hipStream_t stream) {
    const float* src  = (const float*)d_in[0];   // [2,2,128,160,128] f32
    const float* flow = (const float*)d_in[1];   // [2,3,128,160,128] f32
    float* out = (float*)d_out;                  // [2,2,128,160,128] f32

    dim3 grid(Hn / 8, Dn, Bn);                   // 20 x 128 x 2 = 5120 blocks
    dim3 block(256);                             // 8 wave32 per block

    SpatialTransformer_warp3d_kernel<<<grid, block, 0, stream>>>(src, flow, out);
}